// StabilityLossCoP_80041010528546
// MI455X (gfx1250) — compile-verified
//
#include <hip/hip_runtime.h>
#include <hip/hip_bf16.h>
#include <math.h>

// Problem constants (fixed by the reference's setup_inputs)
#define B_   256
#define V_   6890
#define F_   13776
#define N_   20000
#define P_   10
#define NV_  689
#define NF_  1377
#define LB_  32
#define SLAB_ (V_ * 3)          // 20670 f32 = 82,680 bytes per-batch vertex slab

#define COP_W_ 10.0f
#define COP_K_ 100.0f
#define EPS_   1e-6f

typedef float v2f __attribute__((ext_vector_type(2)));
typedef float v8f __attribute__((ext_vector_type(8)));
typedef unsigned int v4u __attribute__((ext_vector_type(4)));
typedef int v8i __attribute__((ext_vector_type(8)));
typedef int v4i __attribute__((ext_vector_type(4)));

// ---------------------------------------------------------------------------
// TDM: bulk-copy a contiguous run of f32 from global memory into LDS.
// 1-row 2D descriptor: tile_dim0 = nelem (fits 16 bits for 20670), data_size=4B.
// Issued by one wave (EXEC is ignored by tensor ops); that wave then waits on
// TENSORcnt before the workgroup barrier.
// ---------------------------------------------------------------------------
__device__ inline void tdm_load_f32_1d(unsigned lds_addr, const void* gsrc, unsigned nelem) {
  unsigned long long ga = (unsigned long long)(uintptr_t)gsrc;
  v4u g0;
  g0.x = 1u;                                            // count=1 (valid descriptor)
  g0.y = lds_addr;                                      // LDS byte address
  g0.z = (unsigned)(ga & 0xFFFFFFFFu);                  // global_addr[31:0]
  g0.w = (unsigned)((ga >> 32) & 0x01FFFFFFu) | (2u << 30); // global_addr[56:32] | type=2
  v8i g1;
  g1[0] = (int)(2u << 16);                              // wg_mask=0, data_size=2 (4 bytes)
  g1[1] = (int)((nelem & 0xFFFFu) << 16);               // tensor_dim0[15:0] (bits 63:48)
  g1[2] = (int)((nelem >> 16) & 0xFFFFu) | (1 << 16);   // tensor_dim0[31:16] | tensor_dim1=1
  g1[3] = (int)((nelem & 0xFFFFu) << 16);               // tensor_dim1 hi=0 | tile_dim0=nelem
  g1[4] = 1;                                            // tile_dim1=1, tile_dim2=0
  g1[5] = (int)nelem;                                   // tensor_dim0_stride[31:0]
  g1[6] = 0;                                            // stride hi / dim1_stride lo
  g1[7] = 0;
  v4i z4 = {0, 0, 0, 0};
#if __clang_major__ >= 23
  v8i z8 = {0, 0, 0, 0, 0, 0, 0, 0};
  __builtin_amdgcn_tensor_load_to_lds(g0, g1, z4, z4, z8, 0);
#else
  __builtin_amdgcn_tensor_load_to_lds(g0, g1, z4, z4, 0);
#endif
}

// Wave32 full-sum (all lanes receive the total).
__device__ inline float wave_sum(float v) {
  v += __shfl_xor(v, 16, 32);
  v += __shfl_xor(v, 8, 32);
  v += __shfl_xor(v, 4, 32);
  v += __shfl_xor(v, 2, 32);
  v += __shfl_xor(v, 1, 32);
  return v;
}

// ---------------------------------------------------------------------------
// WMMA column-sum reduction: partials is an 8x16 row-major [wave][quantity]
// array in LDS. With A == all-ones(16x4), V_WMMA_F32_16X16X4_F32 computes
// D[m,n] = sum_k B[k,n]; two chained WMMAs (C accumulate) cover all 8 waves.
// Exact fp32. Must be called by a fully-active wave (EXEC all ones).
// Returns, in every lane L, the total for quantity (L % 16).
// ---------------------------------------------------------------------------
__device__ inline float wmma_colsum_8x16(const float* p) {
  const int lane = threadIdx.x & 31;
  const int n = lane & 15;
  const int kb = (lane < 16) ? 0 : 2;   // which K rows this lane's B VGPRs carry
  v2f a; a.x = 1.0f; a.y = 1.0f;        // all-ones A (layout-independent)
  v8f c = {};
  v2f b0; b0.x = p[(kb + 0) * 16 + n]; b0.y = p[(kb + 1) * 16 + n]; // waves 0..3
  c = __builtin_amdgcn_wmma_f32_16x16x4_f32(false, a, false, b0, (short)0, c, false, false);
  v2f b1; b1.x = p[(kb + 4) * 16 + n]; b1.y = p[(kb + 5) * 16 + n]; // waves 4..7
  c = __builtin_amdgcn_wmma_f32_16x16x4_f32(false, a, false, b1, (short)0, c, false, false);
  return c[0];                           // every D row equals the column sum
}

// ---------------------------------------------------------------------------
// Kernel A: per-(batch, part) signed volume. One block per batch, 8 waves.
// ---------------------------------------------------------------------------
__global__ __launch_bounds__(256) void vol_kernel(
    const float* __restrict__ vertices, const int* __restrict__ faces,
    const int* __restrict__ part_vids, const int* __restrict__ part_fids,
    const int* __restrict__ boundary_vids, float* __restrict__ vol) {
  __shared__ float slab[SLAB_];
  __shared__ float mv[P_][3];     // boundary mean vertex (the appended vertex V)
  __shared__ float pm[P_][3];     // part mean
  __shared__ float partials[8][16];

  const int b = blockIdx.x;
  const int tid = threadIdx.x;
  const int wave = tid >> 5;
  const int lane = tid & 31;

  if (tid < 32) {
    tdm_load_f32_1d((unsigned)(uintptr_t)(void*)slab,
                    vertices + (size_t)b * SLAB_, (unsigned)SLAB_);
    __builtin_amdgcn_s_wait_tensorcnt(0);
  }
  if (tid < 128) partials[tid >> 4][tid & 15] = 0.0f;
  __syncthreads();

  // Per-part means: one wave per part (parts 8,9 on second pass).
  for (int p = wave; p < P_; p += 8) {
    const int vid = boundary_vids[p * LB_ + lane];
    const float mx = wave_sum(slab[vid * 3 + 0]) * (1.0f / LB_);
    const float my = wave_sum(slab[vid * 3 + 1]) * (1.0f / LB_);
    const float mz = wave_sum(slab[vid * 3 + 2]) * (1.0f / LB_);
    float sx = 0.0f, sy = 0.0f, sz = 0.0f;
    for (int i = lane; i < NV_; i += 32) {
      const int id = part_vids[p * NV_ + i];
      sx += slab[id * 3 + 0]; sy += slab[id * 3 + 1]; sz += slab[id * 3 + 2];
    }
    sx = wave_sum(sx); sy = wave_sum(sy); sz = wave_sum(sz);
    if (lane == 0) {
      mv[p][0] = mx; mv[p][1] = my; mv[p][2] = mz;
      pm[p][0] = (sx + mx) * (1.0f / (NV_ + 1));
      pm[p][1] = (sy + my) * (1.0f / (NV_ + 1));
      pm[p][2] = (sz + mz) * (1.0f / (NV_ + 1));
    }
  }
  __syncthreads();

  // Determinant accumulation: all 10 parts kept in registers per thread.
  float acc[P_];
#pragma unroll
  for (int p = 0; p < P_; ++p) acc[p] = 0.0f;

  for (int p = 0; p < P_; ++p) {
    const float pmx = pm[p][0], pmy = pm[p][1], pmz = pm[p][2];
    const float mvx = mv[p][0], mvy = mv[p][1], mvz = mv[p][2];
    float s = 0.0f;
    for (int j = tid; j < NF_ + (LB_ - 1); j += 256) {
      int i0, i1;
      float cx, cy, cz;
      if (j < NF_) {
        const int fid = part_fids[p * NF_ + j];
        i0 = faces[fid * 3 + 0];
        i1 = faces[fid * 3 + 1];
        const int i2 = faces[fid * 3 + 2];
        cx = slab[i2 * 3 + 0]; cy = slab[i2 * 3 + 1]; cz = slab[i2 * 3 + 2];
      } else {
        const int j2 = j - NF_;
        i0 = boundary_vids[p * LB_ + j2 + 1];
        i1 = boundary_vids[p * LB_ + j2];
        cx = mvx; cy = mvy; cz = mvz;        // third vertex is appended mean (idx V)
      }
      const float ax = slab[i0 * 3 + 0] - pmx, ay = slab[i0 * 3 + 1] - pmy, az = slab[i0 * 3 + 2] - pmz;
      const float bx = slab[i1 * 3 + 0] - pmx, by = slab[i1 * 3 + 1] - pmy, bz = slab[i1 * 3 + 2] - pmz;
      cx -= pmx; cy -= pmy; cz -= pmz;
      // det(rows a,b,c) = a . (b x c)  == reference's 6-term expansion
      s += ax * (by * cz - bz * cy) - ay * (bx * cz - bz * cx) + az * (bx * cy - by * cx);
    }
    acc[p] = s;
  }

#pragma unroll
  for (int p = 0; p < P_; ++p) {
    const float s = wave_sum(acc[p]);
    if (lane == 0) partials[wave][p] = s;
  }
  __syncthreads();

  if (tid < 32) {
    const float tot = wmma_colsum_8x16(&partials[0][0]);  // lane q holds part q total
    if (lane < P_) vol[b * P_ + lane] = fabsf(tot) * (1.0f / 6.0f);
  }
}

// ---------------------------------------------------------------------------
// Kernel B: HD interpolation + COM/CoP accumulators. One block per batch.
// Quantities: 0..2 com_num, 3 com_den, 4..6 cop_num, 7 cop_den.
// ---------------------------------------------------------------------------
__global__ __launch_bounds__(256) void hd_kernel(
    const float* __restrict__ vertices, const float* __restrict__ hd_vals,
    const int* __restrict__ hd_cols, const int* __restrict__ hd_part,
    const float* __restrict__ vol,
    float* __restrict__ com_num, float* __restrict__ com_den,
    float* __restrict__ cop_num, float* __restrict__ cop_den) {
  __shared__ float slab[SLAB_];
  __shared__ float partials[8][16];

  const int b = blockIdx.x;
  const int tid = threadIdx.x;
  const int wave = tid >> 5;
  const int lane = tid & 31;

  if (tid < 32) {
    tdm_load_f32_1d((unsigned)(uintptr_t)(void*)slab,
                    vertices + (size_t)b * SLAB_, (unsigned)SLAB_);
    __builtin_amdgcn_s_wait_tensorcnt(0);
  }
  if (tid < 128) partials[tid >> 4][tid & 15] = 0.0f;
  __syncthreads();

  float cn0 = 0.f, cn1 = 0.f, cn2 = 0.f, cd = 0.f;
  float pn0 = 0.f, pn1 = 0.f, pn2 = 0.f, pd = 0.f;

  for (int n = tid; n < N_; n += 256) {
    const int c0 = hd_cols[n * 3 + 0];
    const int c1 = hd_cols[n * 3 + 1];
    const int c2 = hd_cols[n * 3 + 2];
    const float w0 = hd_vals[n * 3 + 0];
    const float w1 = hd_vals[n * 3 + 1];
    const float w2 = hd_vals[n * 3 + 2];
    const float hx = w0 * slab[c0 * 3 + 0] + w1 * slab[c1 * 3 + 0] + w2 * slab[c2 * 3 + 0];
    const float hy = w0 * slab[c0 * 3 + 1] + w1 * slab[c1 * 3 + 1] + w2 * slab[c2 * 3 + 1];
    const float hz = w0 * slab[c0 * 3 + 2] + w1 * slab[c1 * 3 + 2] + w2 * slab[c2 * 3 + 2];
    const float wv = vol[b * P_ + hd_part[n]];
    cn0 += hx * wv; cn1 += hy * wv; cn2 += hz * wv; cd += wv;
    const float pw = (hy < 0.0f) ? fmaf(-COP_K_, hy, 1.0f) : expf(-COP_W_ * hy);
    pn0 += hx * pw; pn1 += hy * pw; pn2 += hz * pw; pd += pw;
  }

  float q[8] = {cn0, cn1, cn2, cd, pn0, pn1, pn2, pd};
#pragma unroll
  for (int i = 0; i < 8; ++i) {
    const float s = wave_sum(q[i]);
    if (lane == 0) partials[wave][i] = s;
  }
  __syncthreads();

  if (tid < 32) {
    const float tot = wmma_colsum_8x16(&partials[0][0]);  // lane i holds quantity i
    if (lane < 3)                 com_num[b * 3 + lane] = tot;
    else if (lane == 3)           com_den[b] = tot;
    else if (lane < 7)            cop_num[b * 3 + (lane - 4)] = tot;
    else if (lane == 7)           cop_den[b] = tot;
  }
}

// ---------------------------------------------------------------------------
// Kernel C: finalize.  out = [loss(256) | com(256x3) | cop(256x3) | com_xz(256x3)]
// ---------------------------------------------------------------------------
__global__ __launch_bounds__(256) void finalize_kernel(
    const float* __restrict__ com_num, const float* __restrict__ com_den,
    const float* __restrict__ cop_num, const float* __restrict__ cop_den,
    float* __restrict__ out) {
  const int b = threadIdx.x;
  const float cd = com_den[b];
  const float pd = cop_den[b] + EPS_;
  const float comx = com_num[b * 3 + 0] / cd;
  const float comy = com_num[b * 3 + 1] / cd;
  const float comz = com_num[b * 3 + 2] / cd;
  const float copx = cop_num[b * 3 + 0] / pd;
  const float copy_ = cop_num[b * 3 + 1] / pd;
  const float copz = cop_num[b * 3 + 2] / pd;
  const float dx = comx - copx;
  const float dz = comz - copz;
  out[b] = sqrtf(dx * dx + dz * dz);
  out[B_ + b * 3 + 0] = comx;
  out[B_ + b * 3 + 1] = comy;
  out[B_ + b * 3 + 2] = comz;
  out[B_ + 3 * B_ + b * 3 + 0] = copx;
  out[B_ + 3 * B_ + b * 3 + 1] = copy_;
  out[B_ + 3 * B_ + b * 3 + 2] = copz;
  out[B_ + 6 * B_ + b * 3 + 0] = comx;
  out[B_ + 6 * B_ + b * 3 + 1] = 0.0f;
  out[B_ + 6 * B_ + b * 3 + 2] = comz;
}

extern "C" void kernel_launch(void* const* d_in, const int* in_sizes, int n_in,
                              void* d_out, int out_size, void* d_ws, size_t ws_size,
                              hipStream_t stream) {
  const float* vertices      = (const float*)d_in[0];
  const float* hd_vals       = (const float*)d_in[1];
  const int*   faces         = (const int*)d_in[2];
  const int*   part_vids     = (const int*)d_in[3];
  const int*   part_fids     = (const int*)d_in[4];
  const int*   boundary_vids = (const int*)d_in[5];
  const int*   hd_cols       = (const int*)d_in[6];
  const int*   hd_part       = (const int*)d_in[7];

  float* ws      = (float*)d_ws;
  float* vol     = ws;           // B*P   = 2560
  float* com_num = ws + 2560;    // B*3   = 768
  float* com_den = ws + 3328;    // B     = 256
  float* cop_num = ws + 3584;    // B*3   = 768
  float* cop_den = ws + 4352;    // B     = 256

  vol_kernel<<<B_, 256, 0, stream>>>(vertices, faces, part_vids, part_fids,
                                     boundary_vids, vol);
  hd_kernel<<<B_, 256, 0, stream>>>(vertices, hd_vals, hd_cols, hd_part, vol,
                                    com_num, com_den, cop_num, cop_den);
  finalize_kernel<<<1, 256, 0, stream>>>(com_num, com_den, cop_num, cop_den,
                                         (float*)d_out);
}